// GraphSAGELayer_19172734010017
// MI455X (gfx1250) — compile-verified
//
#include <hip/hip_runtime.h>
#include <hip/hip_bf16.h>

#define N_NODES 50000
#define N_EDGES 800000
#define DIN 96
#define DOUT 128

typedef __attribute__((ext_vector_type(2))) float v2f;
typedef __attribute__((ext_vector_type(8))) float v8f;

// ---------------- Phase 0: zero the workspace (summed + count) ----------------
__global__ void sage_zero_kernel(float* __restrict__ ws, int n) {
    int i = blockIdx.x * blockDim.x + threadIdx.x;
    int stride = gridDim.x * blockDim.x;
    for (; i < n; i += stride) ws[i] = 0.0f;
}

// ---------------- Phase 1: edge scatter (mean-aggregate numerator + counts) ----
// One wave32 per edge; lane l handles dims l, l+32, l+64 (96 = 3*32).
// Gathers from x[src] (L2-resident, 19.2MB) and atomically accumulates into
// summed[dst] via hardware global_atomic_add_f32.
__global__ __launch_bounds__(256) void sage_scatter_kernel(
    const float* __restrict__ x, const int* __restrict__ ei,
    float* __restrict__ summed, float* __restrict__ count) {
    const int lane = threadIdx.x & 31;
    const int e = blockIdx.x * 8 + (threadIdx.x >> 5);
    if (e >= N_EDGES) return;
    const int src = ei[e];
    const int dst = ei[N_EDGES + e];
    const float* __restrict__ xs = x + (size_t)src * DIN;
    float* __restrict__ sd = summed + (size_t)dst * DIN;
#pragma unroll
    for (int i = 0; i < 3; ++i) {
        unsafeAtomicAdd(&sd[lane + 32 * i], xs[lane + 32 * i]);
    }
    if (lane == 0) unsafeAtomicAdd(&count[dst], 1.0f);
}

// ---------------- Phase 2: fused dual-GEMM + mean-scale + bias + ReLU ---------
// out = relu( (summed/max(count,1)) @ Wl^T + x @ Wr^T + bl )
// Block = 256 threads = 8 waves; block b owns rows [16b,16b+16); wave w owns
// cols [16w,16w+16). Each wave runs 2x24 V_WMMA_F32_16X16X4_F32 accumulations
// with the per-row 1/count scale applied to the accumulator between passes.
__global__ __launch_bounds__(256) void sage_gemm_kernel(
    const float* __restrict__ x, const float* __restrict__ summed,
    const float* __restrict__ count, const float* __restrict__ Wl,
    const float* __restrict__ bl, const float* __restrict__ Wr,
    float* __restrict__ out) {
    const int lane = threadIdx.x & 31;
    const int wave = threadIdx.x >> 5;
    const int lr   = lane & 15;   // M for A-frag, N for B/C/D-frags
    const int half = lane >> 4;   // selects K pair (A/B) and row half (C/D)
    const int i0 = blockIdx.x * 16;
    const int j0 = wave * 16;

    v8f c = {};

    // Pass 1: summed @ Wl^T
    {
        const float* __restrict__ Ap = summed + (size_t)(i0 + lr) * DIN + half * 2;
        const float* __restrict__ Bp = Wl + (size_t)(j0 + lr) * DIN + half * 2;
#pragma unroll
        for (int k = 0; k < DIN / 4; ++k) {
            v2f a = *(const v2f*)(Ap + k * 4);
            v2f b = *(const v2f*)(Bp + k * 4);
            c = __builtin_amdgcn_wmma_f32_16x16x4_f32(
                false, a, false, b, (short)0, c, false, false);
        }
    }

    // Mean normalization: scale accumulator rows by 1/max(count,1).
    // c[r] corresponds to output row i0 + r + 8*half.
#pragma unroll
    for (int r = 0; r < 8; ++r) {
        float cnt = count[i0 + r + 8 * half];
        c[r] *= 1.0f / fmaxf(cnt, 1.0f);
    }

    // Pass 2: + x @ Wr^T (accumulates into the scaled accumulator)
    {
        const float* __restrict__ Ap = x + (size_t)(i0 + lr) * DIN + half * 2;
        const float* __restrict__ Bp = Wr + (size_t)(j0 + lr) * DIN + half * 2;
#pragma unroll
        for (int k = 0; k < DIN / 4; ++k) {
            v2f a = *(const v2f*)(Ap + k * 4);
            v2f b = *(const v2f*)(Bp + k * 4);
            c = __builtin_amdgcn_wmma_f32_16x16x4_f32(
                false, a, false, b, (short)0, c, false, false);
        }
    }

    // Bias (same for every row of the tile column) + ReLU + store.
    const float bias = bl[j0 + lr];
#pragma unroll
    for (int r = 0; r < 8; ++r) {
        float v = c[r] + bias;
        out[(size_t)(i0 + r + 8 * half) * DOUT + j0 + lr] = fmaxf(v, 0.0f);
    }
}

extern "C" void kernel_launch(void* const* d_in, const int* in_sizes, int n_in,
                              void* d_out, int out_size, void* d_ws, size_t ws_size,
                              hipStream_t stream) {
    const float* x  = (const float*)d_in[0];   // [N, 96]
    const int*   ei = (const int*)d_in[1];     // [2, E]
    const float* Wl = (const float*)d_in[2];   // [128, 96]
    const float* bl = (const float*)d_in[3];   // [128]
    const float* Wr = (const float*)d_in[4];   // [128, 96]
    float* out = (float*)d_out;                // [N, 128]

    float* summed = (float*)d_ws;                      // N*96 floats
    float* count  = summed + (size_t)N_NODES * DIN;    // N floats

    const int zn = N_NODES * DIN + N_NODES;
    sage_zero_kernel<<<2048, 256, 0, stream>>>(summed, zn);
    sage_scatter_kernel<<<N_EDGES / 8, 256, 0, stream>>>(x, ei, summed, count);
    sage_gemm_kernel<<<N_NODES / 16, 256, 0, stream>>>(x, summed, count, Wl, bl, Wr, out);
}